// S4Block_90821378441792
// MI455X (gfx1250) — compile-verified
//
#include <hip/hip_runtime.h>
#include <hip/hip_bf16.h>

// ---------------------------------------------------------------------------
// S4 block for MI455X (gfx1250). Shapes fixed by the reference:
//   B=16, H=64, L=16384, N=32, DC=128.
// Strategy:
//   * FFT conv replaced by exact chunked linear-recurrence scan (S4D modes).
//   * All GEMMs via native f32 WMMA (v_wmma_f32_16x16x4_f32), LDS-staged.
//   * GEMM tile staging via Tensor Data Mover (tensor_load_to_lds) with
//     LDS row padding done by the TDM pad_interval/pad_amount fields.
//   * BatchNorm stats via LDS float atomics + global float atomics.
// ---------------------------------------------------------------------------

typedef __attribute__((ext_vector_type(2))) float        v2f;
typedef __attribute__((ext_vector_type(8))) float        v8f;
typedef __attribute__((ext_vector_type(4))) unsigned int u32x4;
typedef __attribute__((ext_vector_type(8))) int          i32x8;
typedef __attribute__((ext_vector_type(4))) int          i32x4;

#define B_    16
#define H_    64
#define L_    16384
#define N_    32
#define DC_   128
#define LC_   1024              // scan chunk length
#define NCH_  16                // chunks per sequence = L/LC
#define NSEQ_ (B_ * H_)         // 1024 sequences
#define BHL_  ((size_t)B_ * H_ * L_)

#define XS_   68                // padded LDS row stride (64 + 4 DWORDs)

__device__ __forceinline__ v8f wmma4(v2f a, v2f b, v8f c) {
  // D = A(16x4,f32) * B(4x16,f32) + C(16x16,f32)
  return __builtin_amdgcn_wmma_f32_16x16x4_f32(
      /*neg_a=*/false, a, /*neg_b=*/false, b,
      /*c_mod=*/(short)0, c, /*reuse_a=*/false, /*reuse_b=*/false);
}

#if __has_builtin(__builtin_amdgcn_tensor_load_to_lds) && \
    __has_builtin(__builtin_amdgcn_s_wait_tensorcnt)
#define USE_TDM 1
#else
#define USE_TDM 0
#endif

#if USE_TDM
// Issue a TDM load of a [64 x 64] f32 tile (row stride `stride0` elements)
// from `gaddr` into LDS at byte offset `ldsb`, padding +4 DWORDs per 64-DWORD
// row so LDS rows land at stride XS_ = 68 floats.
__device__ __forceinline__ void tdm_tile_64(unsigned long long gaddr,
                                            unsigned int ldsb,
                                            unsigned int stride0) {
  // D# group 0: count=1 | lds_addr | global_addr[56:0] | type=2
  u32x4 g0;
  g0.x = 1u;                                   // count=1, user mode
  g0.y = ldsb;                                 // LDS byte address
  g0.z = (unsigned int)(gaddr & 0xffffffffu);  // global_addr[31:0]
  g0.w = (unsigned int)((gaddr >> 32) & 0x1ffffffu) | (2u << 30);  // type=2
  // D# group 1
  i32x8 g1;
  g1.s0 = (int)((2u << 16)      // data_size = 4 bytes
              | (1u << 20)      // pad_enable
              | (5u << 22)      // pad_interval code 5 => every 64 DWORDs
              | (3u << 25));    // pad_amount code 3 => 4 DWORDs
  g1.s1 = (int)(64u << 16);     // tensor_dim0 = 64 (atomic_barrier_addr = 0)
  g1.s2 = (int)(64u << 16);     // tensor_dim0[31:16]=0 | tensor_dim1 = 64
  g1.s3 = (int)(64u << 16);     // tensor_dim1[31:16]=0 | tile_dim0 = 64
  g1.s4 = (int)64u;             // tile_dim1 = 64 | tile_dim2 = 0
  g1.s5 = (int)stride0;         // tensor_dim0_stride[31:0]
  g1.s6 = 0;                    // stride0[47:32]=0 | tensor_dim1_stride lo = 0
  g1.s7 = 0;
  i32x4 gz4 = {0, 0, 0, 0};     // groups 2/3: dims zero => unused (2D tensor)
  i32x8 gz8 = {0, 0, 0, 0, 0, 0, 0, 0};
  __builtin_amdgcn_tensor_load_to_lds(g0, g1, gz4, gz4, gz8, 0);
}
#endif

// ---------------------------------------------------------------------------
// Zero the BN statistic accumulators (ws is poisoned by the harness).
__global__ void k_zero(float* stat) {
  int t = threadIdx.x;
  if (t < 128) stat[t] = 0.0f;
}

// ---------------------------------------------------------------------------
// Per-(h,n) S4D coefficients: w = exp(dt*A), Cd = C*(w-1)/A, wL = w^LC.
__global__ void k_coef(const float* __restrict__ log_dt,
                       const float* __restrict__ logAr,
                       const float* __restrict__ Aim,
                       const float* __restrict__ Cre,
                       const float* __restrict__ Cim,
                       float* __restrict__ cf) {
  int t = blockIdx.x * blockDim.x + threadIdx.x;
  if (t >= H_ * N_) return;
  int h = t / N_;
  float dt = __expf(log_dt[h]);
  float ar = -__expf(logAr[t]);
  float ai = Aim[t];
  float dr = ar * dt, di = ai * dt;
  float er = __expf(dr);
  float wr = er * __cosf(di);
  float wi = er * __sinf(di);
  // (w - 1) / A
  float inv = 1.0f / (ar * ar + ai * ai);
  float tr = ((wr - 1.0f) * ar + wi * ai) * inv;
  float ti = (wi * ar - (wr - 1.0f) * ai) * inv;
  float cr = Cre[t] * tr - Cim[t] * ti;
  float ci = Cre[t] * ti + Cim[t] * tr;
  // w^LC (LC = 1024 = 2^10) via 10 complex squarings
  float pr = wr, pi = wi;
#pragma unroll
  for (int i = 0; i < 10; ++i) {
    float nr = pr * pr - pi * pi;
    float ni = 2.0f * pr * pi;
    pr = nr; pi = ni;
  }
  cf[0 * H_ * N_ + t] = wr;
  cf[1 * H_ * N_ + t] = wi;
  cf[2 * H_ * N_ + t] = cr;
  cf[3 * H_ * N_ + t] = ci;
  cf[4 * H_ * N_ + t] = pr;
  cf[5 * H_ * N_ + t] = pi;
}

// ---------------------------------------------------------------------------
// FiLM: gb[b, j] = sum_d cond[b,d] * film_w[j,d] + film_b[j].
// One block, 8 waves; wave w computes the 16x16 tile at columns j0 = 16*w.
__global__ void k_film(const float* __restrict__ cond,
                       const float* __restrict__ fw,
                       const float* __restrict__ fb,
                       float* __restrict__ gb) {
  int tid = threadIdx.x;
  int lane = tid & 31, wave = tid >> 5;
  int ml = lane & 15;   // M row for A, N col for B/D
  int hi = lane >> 4;   // half-wave selects K pair / D row offset
  int j0 = wave * 16;
  v8f acc = {};
#pragma unroll 4
  for (int k = 0; k < DC_; k += 4) {
    v2f a, bm;
    a.x  = cond[ml * DC_ + k + 2 * hi];
    a.y  = cond[ml * DC_ + k + 2 * hi + 1];
    bm.x = fw[(j0 + ml) * DC_ + k + 2 * hi];
    bm.y = fw[(j0 + ml) * DC_ + k + 2 * hi + 1];
    acc = wmma4(a, bm, acc);
  }
#pragma unroll
  for (int r = 0; r < 8; ++r) {
    int row = r + 8 * hi;                       // batch index
    gb[row * (2 * H_) + j0 + ml] = acc[r] + fb[j0 + ml];
  }
}

// ---------------------------------------------------------------------------
// 64x64 channel GEMM over positions: dst[b,g,l] = sum_h W[g,h]*src[b,h,l] (+bias).
// Block = 128 threads (4 waves). Each block: one batch b, 64 positions.
// Tiles staged into LDS by the Tensor Data Mover (padded rows, stride 68).
template <bool DO_PRELU, bool DO_STATS>
__global__ void __launch_bounds__(128)
k_gemm64(const float* __restrict__ Wm, const float* __restrict__ bias,
         const float* __restrict__ src, float* __restrict__ dst,
         const float* __restrict__ alpha, float* gsum, float* gsq) {
  __shared__ __align__(16) float sW[64 * XS_];  // W tile, padded rows
  __shared__ __align__(16) float sX[64 * XS_];  // src tile [64 ch][64 pos]
  __shared__ float lsum[64];
  __shared__ float lsq[64];

  int tid = threadIdx.x;
  int b  = blockIdx.x >> 8;          // L/64 = 256 tiles per batch
  int lt = blockIdx.x & 255;
  int l0 = lt * 64;

  if (DO_STATS && tid < 64) { lsum[tid] = 0.0f; lsq[tid] = 0.0f; }

  const float* xb = src + (size_t)b * H_ * L_;
#if USE_TDM
  if (tid == 0) {
    tdm_tile_64((unsigned long long)(uintptr_t)Wm,
                (unsigned int)(uintptr_t)&sW[0], 64u);
    tdm_tile_64((unsigned long long)(uintptr_t)(xb + l0),
                (unsigned int)(uintptr_t)&sX[0], (unsigned int)L_);
    __builtin_amdgcn_s_wait_tensorcnt(0);
  }
#else
  const float4* wv = reinterpret_cast<const float4*>(Wm);
#pragma unroll
  for (int v = tid; v < 64 * 16; v += 128) {
    int row = v >> 4, c4 = v & 15;
    float4 dw = wv[row * 16 + c4];
    *reinterpret_cast<float4*>(&sW[row * XS_ + c4 * 4]) = dw;
    float4 dx = *reinterpret_cast<const float4*>(xb + (size_t)row * L_ + l0 + c4 * 4);
    *reinterpret_cast<float4*>(&sX[row * XS_ + c4 * 4]) = dx;
  }
#endif
  __syncthreads();

  int lane = tid & 31, wave = tid >> 5;
  int ml = lane & 15;      // A row (within tile) & B/D column
  int hi = lane >> 4;
  int p0 = wave * 16;      // this wave's position tile

  v8f acc[4] = {};
#pragma unroll
  for (int k = 0; k < 64; k += 4) {
    v2f bv;
    bv.x = sX[(k + 2 * hi) * XS_ + p0 + ml];
    bv.y = sX[(k + 2 * hi + 1) * XS_ + p0 + ml];
#pragma unroll
    for (int m = 0; m < 4; ++m) {
      v2f av;
      av.x = sW[(m * 16 + ml) * XS_ + k + 2 * hi];
      av.y = sW[(m * 16 + ml) * XS_ + k + 2 * hi + 1];
      acc[m] = wmma4(av, bv, acc[m]);
    }
  }

  float al = DO_PRELU ? alpha[0] : 0.0f;
  float* db = dst + (size_t)b * H_ * L_ + l0 + p0 + ml;
#pragma unroll
  for (int m = 0; m < 4; ++m) {
#pragma unroll
    for (int r = 0; r < 8; ++r) {
      int g = m * 16 + r + 8 * hi;              // output channel
      float v = acc[m][r] + bias[g];
      if (DO_PRELU) v = v >= 0.0f ? v : al * v;
      db[(size_t)g * L_] = v;
      if (DO_STATS) {
        atomicAdd(&lsum[g], v);
        atomicAdd(&lsq[g], v * v);
      }
    }
  }
  if (DO_STATS) {
    __syncthreads();
    if (tid < 64) {
      atomicAdd(&gsum[tid], lsum[tid]);
      atomicAdd(&gsq[tid], lsq[tid]);
    }
  }
}

// ---------------------------------------------------------------------------
// Scan pass 1: per-chunk local end-states with zero init.
// Thread t handles (seq = t/16, chunk = t%16); 32 complex states in registers.
__global__ void __launch_bounds__(256)
k_pass1(const float* __restrict__ hbuf, const float* __restrict__ cf,
        float2* __restrict__ send) {
  int t = blockIdx.x * 256 + threadIdx.x;       // 0 .. NSEQ*NCH-1
  int seq = t >> 4, chunk = t & (NCH_ - 1);
  int ch = seq & (H_ - 1);
  float wr[N_], wi[N_], sr[N_], si[N_];
#pragma unroll
  for (int n = 0; n < N_; ++n) {
    wr[n] = cf[0 * H_ * N_ + ch * N_ + n];
    wi[n] = cf[1 * H_ * N_ + ch * N_ + n];
    sr[n] = 0.0f; si[n] = 0.0f;
  }
  const float* u = hbuf + (size_t)seq * L_ + (size_t)chunk * LC_;
  for (int jj = 0; jj < LC_; jj += 16) {
    if (jj + 128 < LC_) __builtin_prefetch(u + jj + 128, 0, 0);
#pragma unroll 1
    for (int j = jj; j < jj + 16; ++j) {
      float uv = u[j];
#pragma unroll
      for (int n = 0; n < N_; ++n) {
        float a = sr[n], bb = si[n];
        sr[n] = fmaf(wr[n], a, fmaf(-wi[n], bb, uv));
        si[n] = fmaf(wr[n], bb, wi[n] * a);
      }
    }
  }
  float2* o = send + (size_t)t * N_;
#pragma unroll
  for (int n = 0; n < N_; ++n) o[n] = make_float2(sr[n], si[n]);
}

// ---------------------------------------------------------------------------
// Carry scan across chunks: s_in[c] = wL * s_in[c-1] + s_end[c-1].
__global__ void __launch_bounds__(256)
k_scan(const float* __restrict__ cf, const float2* __restrict__ send,
       float2* __restrict__ sin_) {
  int t = blockIdx.x * 256 + threadIdx.x;       // 0 .. NSEQ*N-1
  int seq = t >> 5, n = t & 31;
  int ch = seq & (H_ - 1);
  float wlr = cf[4 * H_ * N_ + ch * N_ + n];
  float wli = cf[5 * H_ * N_ + ch * N_ + n];
  float cr = 0.0f, ci = 0.0f;
#pragma unroll
  for (int c = 0; c < NCH_; ++c) {
    size_t idx = ((size_t)seq * NCH_ + c) * N_ + n;
    sin_[idx] = make_float2(cr, ci);
    float2 e = send[idx];
    float nr = fmaf(wlr, cr, fmaf(-wli, ci, e.x));
    float ni = fmaf(wlr, ci, fmaf(wli, cr, e.y));
    cr = nr; ci = ni;
  }
}

// ---------------------------------------------------------------------------
// Scan pass 2: recurrence with correct initial state, emit
// y[l] = 2*Re(sum_n Cd_n s_n[l]) + Dskip*u[l], written in-place over h.
__global__ void __launch_bounds__(256)
k_pass2(float* __restrict__ hbuf, const float* __restrict__ cf,
        const float2* __restrict__ sin_, const float* __restrict__ dskip) {
  int t = blockIdx.x * 256 + threadIdx.x;
  int seq = t >> 4, chunk = t & (NCH_ - 1);
  int ch = seq & (H_ - 1);
  float wr[N_], wi[N_], cr[N_], ci[N_], sr[N_], si[N_];
  const float2* s0 = sin_ + (size_t)t * N_;
#pragma unroll
  for (int n = 0; n < N_; ++n) {
    wr[n] = cf[0 * H_ * N_ + ch * N_ + n];
    wi[n] = cf[1 * H_ * N_ + ch * N_ + n];
    cr[n] = cf[2 * H_ * N_ + ch * N_ + n];
    ci[n] = cf[3 * H_ * N_ + ch * N_ + n];
    float2 s = s0[n];
    sr[n] = s.x; si[n] = s.y;
  }
  float dk = dskip[ch];
  float* u = hbuf + (size_t)seq * L_ + (size_t)chunk * LC_;
  for (int jj = 0; jj < LC_; jj += 16) {
    if (jj + 128 < LC_) __builtin_prefetch(u + jj + 128, 0, 0);
#pragma unroll 1
    for (int j = jj; j < jj + 16; ++j) {
      float uv = u[j];
      float acc = 0.0f;
#pragma unroll
      for (int n = 0; n < N_; ++n) {
        float a = sr[n], bb = si[n];
        sr[n] = fmaf(wr[n], a, fmaf(-wi[n], bb, uv));
        si[n] = fmaf(wr[n], bb, wi[n] * a);
        acc = fmaf(cr[n], sr[n], acc);
        acc = fmaf(-ci[n], si[n], acc);
      }
      u[j] = fmaf(2.0f, acc, dk * uv);
    }
  }
}

// ---------------------------------------------------------------------------
// Finalize BN statistics: mean and rstd per channel.
__global__ void k_statfin(float* stat) {
  int g = threadIdx.x;  // 64 threads
  float inv = 1.0f / (float)((size_t)B_ * L_);
  float mean = stat[g] * inv;
  float var  = stat[64 + g] * inv - mean * mean;
  stat[128 + g] = mean;
  stat[192 + g] = rsqrtf(var + 1e-5f);
}

// ---------------------------------------------------------------------------
// Fused BN + FiLM + PReLU + depthwise residual (float4 vectorized).
__global__ void __launch_bounds__(256)
k_final(const float4* __restrict__ y1, const float4* __restrict__ x,
        const float* __restrict__ gb, const float* __restrict__ stat,
        const float* __restrict__ resw, const float* __restrict__ a2p,
        float4* __restrict__ out) {
  size_t v = (size_t)blockIdx.x * blockDim.x + threadIdx.x;  // < BHL/4
  size_t e = v * 4;
  int b = (int)(e / ((size_t)H_ * L_));
  int g = (int)((e / L_) & (H_ - 1));
  float mean = stat[128 + g], rstd = stat[192 + g];
  float gamma = gb[b * 2 * H_ + g];
  float beta  = gb[b * 2 * H_ + H_ + g];
  float rw = resw[g];
  float a2 = a2p[0];
  float4 yv = y1[v];
  float4 xv = x[v];
  float4 o;
  float* yy = &yv.x;
  float* xx = &xv.x;
  float* oo = &o.x;
#pragma unroll
  for (int j = 0; j < 4; ++j) {
    float tt = fmaf(gamma, (yy[j] - mean) * rstd, beta);
    tt = tt >= 0.0f ? tt : a2 * tt;
    oo[j] = fmaf(rw, xx[j], tt);
  }
  out[v] = o;
}

// ---------------------------------------------------------------------------
extern "C" void kernel_launch(void* const* d_in, const int* in_sizes, int n_in,
                              void* d_out, int out_size, void* d_ws, size_t ws_size,
                              hipStream_t stream) {
  (void)in_sizes; (void)n_in; (void)out_size; (void)ws_size;
  const float* x    = (const float*)d_in[0];
  const float* cond = (const float*)d_in[1];
  const float* w1   = (const float*)d_in[2];
  const float* b1   = (const float*)d_in[3];
  const float* a1   = (const float*)d_in[4];
  const float* ldt  = (const float*)d_in[5];
  const float* lAr  = (const float*)d_in[6];
  const float* Aim  = (const float*)d_in[7];
  const float* Cre  = (const float*)d_in[8];
  const float* Cim  = (const float*)d_in[9];
  const float* dsk  = (const float*)d_in[10];
  const float* w2   = (const float*)d_in[11];
  const float* b2   = (const float*)d_in[12];
  const float* fw   = (const float*)d_in[13];
  const float* fb   = (const float*)d_in[14];
  const float* a2   = (const float*)d_in[15];
  const float* rw   = (const float*)d_in[16];

  // Workspace layout (floats):
  float*  ws   = (float*)d_ws;
  float*  buf0 = ws;                                  // h, then y_ssm in-place
  float*  buf1 = ws + BHL_;                           // y1 (pre-BN)
  float*  cf   = buf1 + BHL_;                         // 6 * H * N coefficients
  float2* send = (float2*)(cf + 6 * H_ * N_);         // NSEQ*NCH*N carries
  float2* sin_ = send + (size_t)NSEQ_ * NCH_ * N_;    // incoming states
  float*  gb   = (float*)(sin_ + (size_t)NSEQ_ * NCH_ * N_);  // B x 2H FiLM
  float*  stat = gb + B_ * 2 * H_;                    // sum|sq|mean|rstd (4x64)

  k_zero<<<1, 128, 0, stream>>>(stat);
  k_coef<<<(H_ * N_ + 255) / 256, 256, 0, stream>>>(ldt, lAr, Aim, Cre, Cim, cf);
  k_film<<<1, 256, 0, stream>>>(cond, fw, fb, gb);

  // h = prelu(W1 @ x + b1)
  k_gemm64<true, false><<<B_ * (L_ / 64), 128, 0, stream>>>(
      w1, b1, x, buf0, a1, nullptr, nullptr);

  // S4D chunked scan (exact replacement for the FFT convolution)
  k_pass1<<<NSEQ_ * NCH_ / 256, 256, 0, stream>>>(buf0, cf, send);
  k_scan<<<NSEQ_ * N_ / 256, 256, 0, stream>>>(cf, send, sin_);
  k_pass2<<<NSEQ_ * NCH_ / 256, 256, 0, stream>>>(buf0, cf, sin_, dsk);

  // y1 = W2 @ y_ssm + b2, with BN statistics
  k_gemm64<false, true><<<B_ * (L_ / 64), 128, 0, stream>>>(
      w2, b2, buf0, buf1, a1, stat, stat + 64);

  k_statfin<<<1, 64, 0, stream>>>(stat);

  k_final<<<(unsigned)(BHL_ / 4 / 256), 256, 0, stream>>>(
      (const float4*)buf1, (const float4*)x, gb, stat, rw, a2, (float4*)d_out);
}